// Student_LightGCL_73890617360938
// MI455X (gfx1250) — compile-verified
//
#include <hip/hip_runtime.h>

// ---------------------------------------------------------------------------
// LightGCL 2-layer bipartite propagation, COO edge-scatter formulation.
//   Zu[r] += v * Ei[c];  Zi[c] += v * Eu[r]   for each edge (v, r, c)
// Memory-bound (AI ~0.4 FLOP/B). Gather rows coalesced (wave32 x float2 =
// 256B row), scatter via L2-resident native f32 atomics (unsafeAtomicAdd ->
// global_atomic_add_f32), edge metadata staged to LDS with gfx1250 async
// loads (ASYNCcnt double-buffer), gather rows warmed with global_prefetch_b8.
// ---------------------------------------------------------------------------

#define EMB_D 64
constexpr int CHUNK = 128;   // edges per wave-chunk
constexpr int WAVES = 8;     // waves per block (256 threads, wave32)
constexpr int BLOCK = WAVES * 32;

#if defined(__HIP_DEVICE_COMPILE__) && __has_builtin(__builtin_amdgcn_global_load_async_to_lds_b128)
#define HAVE_ASYNC 1
#else
#define HAVE_ASYNC 0
#endif

#if defined(__HIP_DEVICE_COMPILE__) && __has_builtin(__builtin_amdgcn_s_wait_asynccnt)
#define WAIT_ASYNC(n) __builtin_amdgcn_s_wait_asynccnt(n)
#elif HAVE_ASYNC
#define WAIT_ASYNC(n) asm volatile("s_wait_asynccnt " #n ::: "memory")
#else
#define WAIT_ASYNC(n) ((void)0)
#endif

#if HAVE_ASYNC
// Builtin params are pointers to int-vector(16B). Qualify with the segment
// address spaces; generic is a superset so these convert either way.
typedef int v4i_t __attribute__((vector_size(16)));
typedef __attribute__((address_space(1))) v4i_t* gas_v4i_ptr;
typedef __attribute__((address_space(3))) v4i_t* las_v4i_ptr;
#endif

// Native hardware f32 atomic add (coarse-grained device memory — matches the
// reference's order-free segment_sum accumulation).
__device__ __forceinline__ void atom_add_f32(float* p, float v)
{
    unsafeAtomicAdd(p, v);
}

// Stage a full 128-edge chunk of (vals, rows, cols) into per-wave LDS.
// Each lane moves 16B (4 edges) of each array -> 3 async b128 ops per wave.
__device__ __forceinline__ void stage_chunk_async(
    const float* __restrict__ gv, const int* __restrict__ gr,
    const int* __restrict__ gc, long long base, int lane,
    float* sv, int* sr, int* sc)
{
#if HAVE_ASYNC
    __builtin_amdgcn_global_load_async_to_lds_b128(
        (gas_v4i_ptr)(gv + base + lane * 4), (las_v4i_ptr)(sv + lane * 4), 0, 0);
    __builtin_amdgcn_global_load_async_to_lds_b128(
        (gas_v4i_ptr)(gr + base + lane * 4), (las_v4i_ptr)(sr + lane * 4), 0, 0);
    __builtin_amdgcn_global_load_async_to_lds_b128(
        (gas_v4i_ptr)(gc + base + lane * 4), (las_v4i_ptr)(sc + lane * 4), 0, 0);
#else
#pragma unroll
    for (int k = 0; k < 4; ++k) {
        int e = lane * 4 + k;
        sv[e] = gv[base + e];
        sr[e] = gr[base + e];
        sc[e] = gc[base + e];
    }
#endif
}

// Bounds-checked tail staging (plain loads; only the final partial chunk).
__device__ __forceinline__ void stage_chunk_plain(
    const float* __restrict__ gv, const int* __restrict__ gr,
    const int* __restrict__ gc, long long base, int cnt, int lane,
    float* sv, int* sr, int* sc)
{
    for (int e = lane; e < cnt; e += 32) {
        sv[e] = gv[base + e];
        sr[e] = gr[base + e];
        sc[e] = gc[base + e];
    }
}

__global__ __launch_bounds__(BLOCK) void lightgcl_spmm_coo(
    const float* __restrict__ vals, const int* __restrict__ rows,
    const int* __restrict__ cols,
    const float* __restrict__ srcU, const float* __restrict__ srcI,
    float* __restrict__ dstU, float* __restrict__ dstI, int nnz)
{
    __shared__ float sV[WAVES][2][CHUNK];
    __shared__ int   sR[WAVES][2][CHUNK];
    __shared__ int   sC[WAVES][2][CHUNK];

    const int lane = threadIdx.x & 31;
    const int wv   = threadIdx.x >> 5;
    const long long stride = (long long)gridDim.x * WAVES * CHUNK;
    long long start = ((long long)blockIdx.x * WAVES + wv) * CHUNK;
    if (start >= nnz) return;

    int  buf = 0;
    bool cur_async = (start + CHUNK) <= (long long)nnz;
    if (cur_async)
        stage_chunk_async(vals, rows, cols, start, lane,
                          sV[wv][0], sR[wv][0], sC[wv][0]);
    else
        stage_chunk_plain(vals, rows, cols, start, (int)(nnz - start), lane,
                          sV[wv][0], sR[wv][0], sC[wv][0]);

    while (start < nnz) {
        // Prefetch next chunk's metadata into the other LDS buffer.
        const long long nxt = start + stride;
        bool nxt_async = false;
        if (nxt < nnz) {
            nxt_async = (nxt + CHUNK) <= (long long)nnz;
            if (nxt_async)
                stage_chunk_async(vals, rows, cols, nxt, lane,
                                  sV[wv][buf ^ 1], sR[wv][buf ^ 1], sC[wv][buf ^ 1]);
            else
                stage_chunk_plain(vals, rows, cols, nxt, (int)(nnz - nxt), lane,
                                  sV[wv][buf ^ 1], sR[wv][buf ^ 1], sC[wv][buf ^ 1]);
        }
        // Async loads complete in order: cnt<=3 means this chunk's 3 are done.
        if (cur_async) {
            if (nxt_async) { WAIT_ASYNC(3); } else { WAIT_ASYNC(0); }
        }
        asm volatile("" ::: "memory");

        const int cnt = ((long long)nnz - start >= CHUNK) ? CHUNK
                                                          : (int)(nnz - start);
        const float* svp = sV[wv][buf];
        const int*   srp = sR[wv][buf];
        const int*   scp = sC[wv][buf];

        // Prefetch the gather rows for this chunk (global_prefetch_b8);
        // edge e's actual loads happen up to CHUNK wave-iterations later.
#pragma unroll
        for (int k = 0; k < CHUNK / 32; ++k) {
            int e = k * 32 + lane;
            if (e < cnt) {
                __builtin_prefetch(srcI + ((size_t)scp[e] << 6), 0, 0);
                __builtin_prefetch(srcU + ((size_t)srp[e] << 6), 0, 0);
            }
        }

        // One edge per wave-iteration: 32 lanes x float2 covers D=64 exactly
        // (one fully coalesced 256B row per load).
        for (int e = 0; e < cnt; ++e) {
            const float v = svp[e];
            const int   r = srp[e];
            const int   c = scp[e];
            const float2 ei =
                *(reinterpret_cast<const float2*>(srcI + ((size_t)c << 6)) + lane);
            const float2 eu =
                *(reinterpret_cast<const float2*>(srcU + ((size_t)r << 6)) + lane);
            float* du = dstU + ((size_t)r << 6) + 2 * lane;
            float* di = dstI + ((size_t)c << 6) + 2 * lane;
            atom_add_f32(du,     v * ei.x);
            atom_add_f32(du + 1, v * ei.y);
            atom_add_f32(di,     v * eu.x);
            atom_add_f32(di + 1, v * eu.y);
        }

        start = nxt;
        buf ^= 1;
        cur_async = nxt_async;
    }
}

// out = concat(user_emb, item_emb) (layer-0 running sum); zero Z buffers.
__global__ void lightgcl_init(const float4* __restrict__ ue,
                              const float4* __restrict__ ie,
                              float4* __restrict__ out,
                              float4* __restrict__ zu,
                              float4* __restrict__ zi,
                              long long nu4, long long total4)
{
    long long i = (long long)blockIdx.x * blockDim.x + threadIdx.x;
    if (i >= total4) return;
    const float4 z = make_float4(0.f, 0.f, 0.f, 0.f);
    if (i < nu4) {
        out[i] = ue[i];
        zu[i]  = z;
    } else {
        out[i]      = ie[i - nu4];
        zi[i - nu4] = z;
    }
}

// out = (out + Z_layer1) * 1/(L+1)   (out already holds E0 + Z_layer2)
__global__ void lightgcl_final(float4* __restrict__ out,
                               const float4* __restrict__ zu,
                               const float4* __restrict__ zi,
                               long long nu4, long long total4, float inv)
{
    long long i = (long long)blockIdx.x * blockDim.x + threadIdx.x;
    if (i >= total4) return;
    float4 o = out[i];
    float4 z = (i < nu4) ? zu[i] : zi[i - nu4];
    o.x = (o.x + z.x) * inv;
    o.y = (o.y + z.y) * inv;
    o.z = (o.z + z.z) * inv;
    o.w = (o.w + z.w) * inv;
    out[i] = o;
}

extern "C" void kernel_launch(void* const* d_in, const int* in_sizes, int n_in,
                              void* d_out, int out_size, void* d_ws, size_t ws_size,
                              hipStream_t stream)
{
    (void)n_in; (void)out_size; (void)ws_size;
    const float* user_emb = (const float*)d_in[0];
    const float* item_emb = (const float*)d_in[1];
    const float* vals     = (const float*)d_in[2];
    const int*   rows     = (const int*)d_in[3];
    const int*   cols     = (const int*)d_in[4];

    const long long nuD = in_sizes[0];   // N_USERS * 64
    const long long niD = in_sizes[1];   // N_ITEMS * 64
    const int       nnz = in_sizes[2];

    float* out  = (float*)d_out;
    float* outU = out;
    float* outI = out + nuD;
    float* zu   = (float*)d_ws;          // N_USERS x 64
    float* zi   = zu + nuD;              // N_ITEMS x 64

    const long long total4 = (nuD + niD) / 4;
    const long long nu4    = nuD / 4;
    const int tb = 256;
    const int gb = (int)((total4 + tb - 1) / tb);

    lightgcl_init<<<gb, tb, 0, stream>>>(
        (const float4*)user_emb, (const float4*)item_emb,
        (float4*)out, (float4*)zu, (float4*)zi, nu4, total4);

    const long long waveChunks = ((long long)nnz + CHUNK - 1) / CHUNK;
    int blocks = (int)((waveChunks + WAVES - 1) / WAVES);
    if (blocks > 2048) blocks = 2048;   // ~16K waves; stride loop covers rest

    // Layer 1: (E_u0, E_i0) -> (Zu, Zi)
    lightgcl_spmm_coo<<<blocks, BLOCK, 0, stream>>>(
        vals, rows, cols, user_emb, item_emb, zu, zi, nnz);
    // Layer 2: (Zu, Zi) -> atomically accumulate straight into out (E0 + Z2)
    lightgcl_spmm_coo<<<blocks, BLOCK, 0, stream>>>(
        vals, rows, cols, zu, zi, outU, outI, nnz);

    lightgcl_final<<<gb, tb, 0, stream>>>(
        (float4*)out, (const float4*)zu, (const float4*)zi,
        nu4, total4, 1.0f / 3.0f);
}